// CMHSA_37907381354634
// MI455X (gfx1250) — compile-verified
//
#include <hip/hip_runtime.h>
#include <hip/hip_bf16.h>

#define Bx 8
#define Cx 384
#define NHx 6
#define HDx 64
#define Tx 1024

typedef __attribute__((ext_vector_type(2))) float v2f;
typedef __attribute__((ext_vector_type(8))) float v8f;

__device__ __forceinline__ v8f wmma_f32_4(v2f a, v2f b, v8f c) {
  // D(16x16,f32) = A(16x4,f32) x B(4x16,f32) + C
  return __builtin_amdgcn_wmma_f32_16x16x4_f32(
      /*neg_a=*/false, a, /*neg_b=*/false, b,
      /*c_mod=*/(short)0, c, /*reuse_a=*/false, /*reuse_b=*/false);
}

// --- gfx1250 async global->LDS copy (ASYNCcnt-tracked) --------------------
__device__ __forceinline__ uint32_t lds_off32(const void* p) {
  // generic LDS pointer: low 32 bits are the in-allocation LDS byte offset
  return (uint32_t)(uintptr_t)p;
}
__device__ __forceinline__ void async_ld_b128(const void* smem_dst,
                                              const float* gsrc) {
  asm volatile("global_load_async_to_lds_b128 %0, %1, off"
               :
               : "v"(lds_off32(smem_dst)), "v"(gsrc)
               : "memory");
}
__device__ __forceinline__ void wait_async0() {
  asm volatile("s_wait_asynccnt 0" ::: "memory");
}

// ---------------------------------------------------------------------------
// Kernel 1: q/k/v = W[o,c] * x[b,c,t]   (M=384, N=1024, K=384, per b, 3 W's)
// block = 128 (4 waves). Block tile: 64(M) x 64(N), K-panel = 32.
// Panels staged in LDS via global_load_async_to_lds_b128.
// grid = (384/64, 1024/64, B*3)
// ---------------------------------------------------------------------------
__global__ __launch_bounds__(128) void qkv_gemm(
    const float* __restrict__ x,
    const float* __restrict__ Wq, const float* __restrict__ Wk,
    const float* __restrict__ Wv,
    float* __restrict__ qb, float* __restrict__ kbuf, float* __restrict__ vb) {
  __shared__ float As[64][36];  // [o][k]  row stride 144B (16B aligned)
  __shared__ float Bs[32][68];  // [k][t]  row stride 272B (16B aligned)

  const int tid = threadIdx.x;
  const int wave = tid >> 5;
  const int lane = tid & 31;
  const int o0b = blockIdx.x * 64;       // block M base
  const int o0 = o0b + wave * 16;        // wave M tile
  const int t0 = blockIdx.y * 64;        // block N base
  const int b = blockIdx.z / 3, which = blockIdx.z % 3;
  const float* W = (which == 0) ? Wq : (which == 1) ? Wk : Wv;
  float* out = (which == 0) ? qb : (which == 1) ? kbuf : vb;
  const float* xb = x + (size_t)b * Cx * Tx;

  const int m = lane & 15;
  const int kb2 = (lane >> 4) * 2;
  const int rbase = (lane >> 4) * 8;

  v8f acc[4] = {};
  for (int k0 = 0; k0 < Cx; k0 += 32) {
    __syncthreads();  // previous panel fully consumed
#pragma unroll
    for (int it = 0; it < 4; ++it) {
      const int i = tid + 128 * it;
      const int ar = i >> 3, ac = (i & 7) * 4;    // A: 64 rows x 8 chunks
      async_ld_b128(&As[ar][ac], &W[(size_t)(o0b + ar) * Cx + k0 + ac]);
      const int br = i >> 4, bc = (i & 15) * 4;   // B: 32 rows x 16 chunks
      async_ld_b128(&Bs[br][bc], &xb[(size_t)(k0 + br) * Tx + t0 + bc]);
    }
    wait_async0();
    __syncthreads();  // all waves' panel data visible
#pragma unroll
    for (int ks = 0; ks < 8; ++ks) {
      v2f a;
      a.x = As[wave * 16 + m][ks * 4 + kb2];
      a.y = As[wave * 16 + m][ks * 4 + kb2 + 1];
#pragma unroll
      for (int nt = 0; nt < 4; ++nt) {
        v2f bf;
        bf.x = Bs[ks * 4 + kb2][nt * 16 + m];
        bf.y = Bs[ks * 4 + kb2 + 1][nt * 16 + m];
        acc[nt] = wmma_f32_4(a, bf, acc[nt]);
      }
    }
  }
  float* ob = out + (size_t)b * Cx * Tx;
#pragma unroll
  for (int nt = 0; nt < 4; ++nt)
#pragma unroll
    for (int j = 0; j < 8; ++j)
      ob[(size_t)(o0 + rbase + j) * Tx + t0 + nt * 16 + m] = acc[nt][j];
}

// ---------------------------------------------------------------------------
// Kernel 2: column sums of v (one wave per row, lane-strided + shfl reduce)
// ---------------------------------------------------------------------------
__global__ void vsum_kernel(const float* __restrict__ vb,
                            float* __restrict__ vsum) {
  const int row = blockIdx.x * (blockDim.x >> 5) + (threadIdx.x >> 5);
  const int lane = threadIdx.x & 31;
  if (row >= Bx * Cx) return;
  const float* p = vb + (size_t)row * Tx;
  float s = 0.f;
  for (int t = lane; t < Tx; t += 32) s += p[t];
#pragma unroll
  for (int off = 16; off; off >>= 1) s += __shfl_xor(s, off, 32);
  if (lane == 0) vsum[row] = s;
}

// ---------------------------------------------------------------------------
// Kernel 3: fused cross-head scores + mix + online softmax (+ sum p^2) + P@V^T
// workgroup = 6 waves (192 thr) = one 16-row q tile, all 6 heads.
// grid = (T/16, B). Produces flash output fo[b,h,q,d] and per-(b,h)
// sum of p^2 into stats via global_atomic_add_f32.
// ---------------------------------------------------------------------------
__global__ __launch_bounds__(192) void attn_kernel(
    const float* __restrict__ qb, const float* __restrict__ kbuf,
    const float* __restrict__ vb, const float* __restrict__ head_w,
    float* __restrict__ fo, float* __restrict__ stats) {
  __shared__ float s_lds[NHx][16][17];   // raw per-head score tiles (padded)
  __shared__ float p_lds[NHx][16][18];   // probs per output head (8B rows)
  __shared__ float corr_lds[NHx][16];
  __shared__ float inv_lds[NHx][16];
  __shared__ float coef[NHx][NHx];

  const int w = threadIdx.x >> 5;        // wave id == head id
  const int lane = threadIdx.x & 31;
  const int b = blockIdx.y;
  const int q0 = blockIdx.x * 16;

  if (threadIdx.x < NHx * NHx) {
    const int g = threadIdx.x / NHx, h = threadIdx.x % NHx;
    coef[g][h] = head_w[g * NHx + h] * 0.125f;  // fold HD^-0.5 = 1/8
  }

  const float* qh = qb + ((size_t)b * Cx + w * HDx) * Tx;
  const float* kh = kbuf + ((size_t)b * Cx + w * HDx) * Tx;
  const float* vh = vb + ((size_t)b * Cx + w * HDx) * Tx;

  const int m = lane & 15;
  const int kb2 = (lane >> 4) * 2;
  const int rbase = (lane >> 4) * 8;

  // hoist q A-fragments for the whole t-loop (K=64 -> 16 wmma steps)
  v2f qa[16];
#pragma unroll
  for (int ks = 0; ks < 16; ++ks) {
    const int d = ks * 4 + kb2;
    qa[ks].x = qh[(size_t)d * Tx + q0 + m];
    qa[ks].y = qh[(size_t)(d + 1) * Tx + q0 + m];
  }

  v8f acc[4] = {};                       // O accumulator: 16q x 64d
  float rm = -3.0e38f, rl = 0.f, rl2 = 0.f;  // per-row state (lanes 0..15)

  __syncthreads();

  for (int tt = 0; tt < Tx / 16; ++tt) {
    const int t0 = tt * 16;

    // ---- phase 1: raw score tile for head w via WMMA ----
    v8f s = {};
#pragma unroll
    for (int ks = 0; ks < 16; ++ks) {
      const int d = ks * 4 + kb2;
      v2f bf;
      bf.x = kh[(size_t)d * Tx + t0 + m];
      bf.y = kh[(size_t)(d + 1) * Tx + t0 + m];
      s = wmma_f32_4(qa[ks], bf, s);
    }
#pragma unroll
    for (int j = 0; j < 8; ++j) s_lds[w][rbase + j][m] = s[j];
    __syncthreads();

    // ---- phase 2: 6x6 head mix + online softmax for output head w ----
    if (lane < 16) {
      const int r = lane;
      float mv[16];
      float mx = -3.0e38f;
#pragma unroll
      for (int c2 = 0; c2 < 16; ++c2) {
        float am = 0.f;
#pragma unroll
        for (int h = 0; h < NHx; ++h) am += coef[w][h] * s_lds[h][r][c2];
        mv[c2] = am;
        mx = fmaxf(mx, am);
      }
      const float newm = fmaxf(rm, mx);
      const float cf = __expf(rm - newm);
      float ls = 0.f, l2s = 0.f;
#pragma unroll
      for (int c2 = 0; c2 < 16; ++c2) {
        const float p = __expf(mv[c2] - newm);
        p_lds[w][r][c2] = p;
        ls += p;
        l2s += p * p;
      }
      rl = rl * cf + ls;
      rl2 = rl2 * cf * cf + l2s;
      rm = newm;
      corr_lds[w][r] = cf;
    }

    // ---- phase 3: rescale O and accumulate P @ V^T (all 32 lanes) ----
    float cfj[8];
#pragma unroll
    for (int j = 0; j < 8; ++j) cfj[j] = corr_lds[w][rbase + j];
#pragma unroll
    for (int nt = 0; nt < 4; ++nt)
#pragma unroll
      for (int j = 0; j < 8; ++j) acc[nt][j] *= cfj[j];

#pragma unroll
    for (int ks = 0; ks < 4; ++ks) {
      const int kk = ks * 4 + kb2;  // local t index of this K-slice
      v2f a;
      a.x = p_lds[w][m][kk];
      a.y = p_lds[w][m][kk + 1];
#pragma unroll
      for (int nt = 0; nt < 4; ++nt) {
        const int d = nt * 16 + m;  // B[k][n] = V^T[t][d] = v[c=w*64+d][t0+k]
        v2f bf;
        bf.x = vh[(size_t)d * Tx + t0 + kk];
        bf.y = vh[(size_t)d * Tx + t0 + kk + 1];
        acc[nt] = wmma_f32_4(a, bf, acc[nt]);
      }
    }
    __syncthreads();
  }

  // ---- finalize: 1/l normalization, variance contribution ----
  if (lane < 16) {
    const float inv = 1.0f / rl;
    inv_lds[w][lane] = inv;
    atomicAdd(&stats[b * NHx + w], rl2 * inv * inv);  // row sum of p^2
  }
  float* foh = fo + (((size_t)b * NHx + w) * Tx + q0) * HDx;
#pragma unroll
  for (int nt = 0; nt < 4; ++nt)
#pragma unroll
    for (int j = 0; j < 8; ++j)
      foh[(size_t)(rbase + j) * HDx + nt * 16 + m] =
          acc[nt][j] * inv_lds[w][rbase + j];
}

// ---------------------------------------------------------------------------
// Kernel 4: InstanceNorm stats -> per (b,head) affine (a, b)
// mean == 1/T exactly (softmax rows sum to 1); var = E[p^2] - mean^2
// ---------------------------------------------------------------------------
__global__ void affine_kernel(const float* __restrict__ stats,
                              const float* __restrict__ gamma,
                              const float* __restrict__ beta,
                              float* __restrict__ aff) {
  const int i = threadIdx.x;
  if (i >= Bx * NHx) return;
  const int g = i % NHx;
  const float meanv = 1.0f / (float)Tx;
  const float ex2 = stats[i] / ((float)Tx * (float)Tx);
  const float rstd = rsqrtf(ex2 - meanv * meanv + 1e-5f);
  const float a = gamma[g] * rstd;
  aff[i * 2 + 0] = a;
  aff[i * 2 + 1] = beta[g] - a * meanv;
}

// ---------------------------------------------------------------------------
// Kernel 5: build pre-projection tensor (reshape is identity on flat memory)
// oscr[b][t][c] = a(b,h)*fo[flat] + b(b,h)*vsum[b, h*64+d]
// ---------------------------------------------------------------------------
__global__ void build_oscr(const float* __restrict__ fo,
                           const float* __restrict__ vsum,
                           const float* __restrict__ aff,
                           float* __restrict__ oscr) {
  const size_t idx = (size_t)blockIdx.x * blockDim.x + threadIdx.x;
  if (idx >= (size_t)Bx * Tx * Cx) return;
  const int b = (int)(idx / ((size_t)Tx * Cx));
  const int f = (int)(idx % ((size_t)Tx * Cx));
  const int h = f / (Tx * HDx);
  const int d = f % HDx;
  const float a = aff[(b * NHx + h) * 2];
  const float bc = aff[(b * NHx + h) * 2 + 1];
  oscr[idx] = a * fo[idx] + bc * vsum[b * Cx + h * HDx + d];
}

// ---------------------------------------------------------------------------
// Kernel 6: y[b][o][t] = projb[o] + sum_ci projW[o][ci]*oscr[b][t][ci]
// LDS-blocked like kernel 1; B panel stored [t][k] (async b128 keeps rows
// contiguous). Bias folded into C-fragment init; writes transposed output.
// ---------------------------------------------------------------------------
__global__ __launch_bounds__(128) void proj_gemm(
    const float* __restrict__ oscr, const float* __restrict__ projW,
    const float* __restrict__ projb, float* __restrict__ out) {
  __shared__ float As[64][36];   // [o][k]
  __shared__ float Bt[64][36];   // [t][k]

  const int tid = threadIdx.x;
  const int wave = tid >> 5;
  const int lane = tid & 31;
  const int o0b = blockIdx.x * 64;
  const int o0 = o0b + wave * 16;
  const int t0 = blockIdx.y * 64;
  const int b = blockIdx.z;
  const float* ob = oscr + (size_t)b * Tx * Cx;

  const int m = lane & 15;
  const int kb2 = (lane >> 4) * 2;
  const int rbase = (lane >> 4) * 8;

  v8f acc[4];
#pragma unroll
  for (int nt = 0; nt < 4; ++nt)
#pragma unroll
    for (int j = 0; j < 8; ++j) acc[nt][j] = projb[o0 + rbase + j];

  for (int k0 = 0; k0 < Cx; k0 += 32) {
    __syncthreads();
#pragma unroll
    for (int it = 0; it < 4; ++it) {
      const int i = tid + 128 * it;
      const int r = i >> 3, c = (i & 7) * 4;  // 64 rows x 8 chunks of 4 floats
      async_ld_b128(&As[r][c], &projW[(size_t)(o0b + r) * Cx + k0 + c]);
      async_ld_b128(&Bt[r][c], &ob[(size_t)(t0 + r) * Cx + k0 + c]);
    }
    wait_async0();
    __syncthreads();
#pragma unroll
    for (int ks = 0; ks < 8; ++ks) {
      v2f a;
      a.x = As[wave * 16 + m][ks * 4 + kb2];
      a.y = As[wave * 16 + m][ks * 4 + kb2 + 1];
#pragma unroll
      for (int nt = 0; nt < 4; ++nt) {
        v2f bf;  // B[k][n] = oscr[t0+n][k]
        bf.x = Bt[nt * 16 + m][ks * 4 + kb2];
        bf.y = Bt[nt * 16 + m][ks * 4 + kb2 + 1];
        acc[nt] = wmma_f32_4(a, bf, acc[nt]);
      }
    }
  }
  float* yb = out + (size_t)b * Cx * Tx;
#pragma unroll
  for (int nt = 0; nt < 4; ++nt)
#pragma unroll
    for (int j = 0; j < 8; ++j)
      yb[(size_t)(o0 + rbase + j) * Tx + t0 + nt * 16 + m] = acc[nt][j];
}

// ---------------------------------------------------------------------------
extern "C" void kernel_launch(void* const* d_in, const int* in_sizes, int n_in,
                              void* d_out, int out_size, void* d_ws, size_t ws_size,
                              hipStream_t stream) {
  const float* x = (const float*)d_in[0];
  const float* Wq = (const float*)d_in[1];
  const float* Wk = (const float*)d_in[2];
  const float* Wv = (const float*)d_in[3];
  const float* head_w = (const float*)d_in[4];
  const float* gamma = (const float*)d_in[5];
  const float* beta = (const float*)d_in[6];
  const float* projW = (const float*)d_in[7];
  const float* projb = (const float*)d_in[8];
  float* out = (float*)d_out;

  const size_t SZ = (size_t)Bx * Cx * Tx * sizeof(float);  // 12.58 MB
  char* ws = (char*)d_ws;
  float* qb   = (float*)(ws + 0 * SZ);
  float* kbuf = (float*)(ws + 1 * SZ);
  float* vb   = (float*)(ws + 2 * SZ);
  float* fo   = (float*)(ws + 3 * SZ);
  float* oscr = (float*)(ws + 4 * SZ);
  float* vsum = (float*)(ws + 5 * SZ);
  float* stats = (float*)(ws + 5 * SZ + (size_t)Bx * Cx * sizeof(float));
  float* aff = stats + Bx * NHx;

  hipMemsetAsync(stats, 0, Bx * NHx * sizeof(float), stream);

  // 1) q/k/v channel GEMMs (async-LDS staged, WMMA f32)
  qkv_gemm<<<dim3(Cx / 64, Tx / 64, Bx * 3), 128, 0, stream>>>(
      x, Wq, Wk, Wv, qb, kbuf, vb);

  // 2) v column sums (wave-per-row)
  vsum_kernel<<<(Bx * Cx) / 8, 256, 0, stream>>>(vb, vsum);

  // 3) fused mixed-head flash attention (+ sum p^2 stats)
  attn_kernel<<<dim3(Tx / 16, Bx), 192, 0, stream>>>(
      qb, kbuf, vb, head_w, fo, stats);

  // 4) norm affine coefficients
  affine_kernel<<<1, 64, 0, stream>>>(stats, gamma, beta, aff);

  // 5) normalized pre-projection tensor (flat reshape == identity)
  {
    const size_t n = (size_t)Bx * Tx * Cx;
    build_oscr<<<(unsigned)((n + 255) / 256), 256, 0, stream>>>(fo, vsum, aff, oscr);
  }

  // 6) output projection + transpose to [B,C,H,W]
  proj_gemm<<<dim3(Cx / 64, Tx / 64, Bx), 128, 0, stream>>>(
      oscr, projW, projb, out);
}